// Decoder_26671746908419
// MI455X (gfx1250) — compile-verified
//
#include <hip/hip_runtime.h>
#include <hip/hip_bf16.h>
#include <math.h>

typedef float v2f __attribute__((ext_vector_type(2)));
typedef float v8f __attribute__((ext_vector_type(8)));

#define NTOK 8192   // B*H1*W1

__device__ __forceinline__ float gelu_exact(float x) {
    return 0.5f * x * (1.0f + erff(x * 0.70710678118654752f));
}

__device__ __forceinline__ v8f wmma_f32(v2f a, v2f b, v8f c) {
    // D = A(16x4,f32) * B(4x16,f32) + C(16x16,f32)
    return __builtin_amdgcn_wmma_f32_16x16x4_f32(false, a, false, b, (short)0, c, false, false);
}

// ---------------------------------------------------------------------------
// Convex upsampling: one block per (n,h,w); thread = ii*8+jj (64 threads)
// ---------------------------------------------------------------------------
__global__ void __launch_bounds__(64)
upsample_kernel(const float* __restrict__ coords1, const float* __restrict__ up_mask,
                float* __restrict__ out)
{
    const int n = blockIdx.x >> 12;
    const int hw = blockIdx.x & 4095;
    const int h = hw >> 6, w = hw & 63;
    __shared__ float pf[2][9];
    const int tid = threadIdx.x;
    if (tid < 18) {
        int c = tid / 9, k = tid % 9;
        int di = k / 3 - 1, dj = k % 3 - 1;
        int hh = h + di, ww = w + dj;
        float v = 0.0f;
        if (hh >= 0 && hh < 64 && ww >= 0 && ww < 64) {
            float cv = coords1[((size_t)n * 2 + c) * 4096 + hh * 64 + ww];
            float base = (c == 0) ? (float)ww : (float)hh;   // flow = coords1 - coords0
            v = 8.0f * (cv - base);
        }
        pf[c][k] = v;
    }
    __syncthreads();
    float m[9]; float mx = -1e30f;
    #pragma unroll
    for (int k = 0; k < 9; ++k) {
        m[k] = up_mask[((size_t)n * 576 + k * 64 + tid) * 4096 + hw];
        mx = fmaxf(mx, m[k]);
    }
    float den = 0.0f;
    #pragma unroll
    for (int k = 0; k < 9; ++k) { m[k] = expf(m[k] - mx); den += m[k]; }
    const float inv = 1.0f / den;
    float o0 = 0.0f, o1 = 0.0f;
    #pragma unroll
    for (int k = 0; k < 9; ++k) { o0 += m[k] * pf[0][k]; o1 += m[k] * pf[1][k]; }
    o0 *= inv; o1 *= inv;
    const int ii = tid >> 3, jj = tid & 7;
    const size_t po = (size_t)(h * 8 + ii) * 512 + (w * 8 + jj);
    out[((size_t)n * 2 + 0) * 262144 + po] = o0;
    out[((size_t)n * 2 + 1) * 262144 + po] = o1;
}

// ---------------------------------------------------------------------------
// encode_flow: 81 bilinear samples per token -> corr[NTOK][84] (cols 81..83 = 0)
// ---------------------------------------------------------------------------
__global__ void __launch_bounds__(256)
corr_kernel(const float* __restrict__ cost_maps, const float* __restrict__ coords1,
            float* __restrict__ corr)
{
    const int gid = blockIdx.x * 256 + threadIdx.x;
    if (gid >= NTOK * 84) return;
    const int token = gid / 84;
    const int j = gid - token * 84;
    if (j >= 81) { corr[gid] = 0.0f; return; }
    const int b = token >> 12;
    const int hw = token & 4095;
    const float cx = coords1[((size_t)b * 2 + 0) * 4096 + hw];
    const float cy = coords1[((size_t)b * 2 + 1) * 4096 + hw];
    const int a = j / 9, bb = j - a * 9;
    // sample x = cx + d[a], y = cy + d[b]  (d = -4..4)
    const float sx = cx + (float)(a - 4);
    const float sy = cy + (float)(bb - 4);
    const float x0 = floorf(sx), y0 = floorf(sy);
    const float wx = sx - x0, wy = sy - y0;
    const float* img = cost_maps + (size_t)token * 4096;
    auto g = [&](float xi, float yi) -> float {
        if (xi < 0.0f || xi > 63.0f || yi < 0.0f || yi > 63.0f) return 0.0f;
        return img[(int)yi * 64 + (int)xi];
    };
    const float v00 = g(x0, y0),        v10 = g(x0 + 1.0f, y0);
    const float v01 = g(x0, y0 + 1.0f), v11 = g(x0 + 1.0f, y0 + 1.0f);
    corr[gid] = v00 * (1.0f - wx) * (1.0f - wy) + v10 * wx * (1.0f - wy)
              + v01 * (1.0f - wx) * wy          + v11 * wx * wy;
}

// ---------------------------------------------------------------------------
// Generic GEMM: out[M x 64] = A[M x KD] @ W + bias, via V_WMMA_F32_16X16X4_F32.
// NK=true: W stored [64][KORIG] (transposed on LDS load, zero-padded to KD).
// MODE: 0 = bias, 1 = bias+GELU, 3 = bias + resid + scatter to NCHW output.
// 256 threads = 8 waves; wave owns one 16-row M tile, N=64 -> 4 accumulators.
// ---------------------------------------------------------------------------
template<int KD, int KORIG, bool NK, int MODE>
__global__ void __launch_bounds__(256)
gemm_n64(const float* __restrict__ A, const float* __restrict__ W,
         const float* __restrict__ bias, const float* __restrict__ resid,
         float* __restrict__ out)
{
    __shared__ float sW[KD * 64];
    for (int i = threadIdx.x; i < KD * 64; i += 256) {
        const int k = i >> 6, n = i & 63;
        float w;
        if (NK) w = (k < KORIG) ? W[n * KORIG + k] : 0.0f;
        else    w = W[i];
        sW[i] = w;
    }
    __syncthreads();
    const int wave = threadIdx.x >> 5;
    const int lane = threadIdx.x & 31;
    const int r16 = lane & 15, kh = lane >> 4;
    const int tile = blockIdx.x * 8 + wave;
    const float* Arow = A + (size_t)(tile * 16 + r16) * KD;
    v8f c0 = {}, c1 = {}, c2 = {}, c3 = {};
    #pragma unroll 4
    for (int kk = 0; kk < KD; kk += 4) {
        v2f a;
        a.x = Arow[kk + kh * 2];
        a.y = Arow[kk + kh * 2 + 1];
        const float* B0 = &sW[(kk + kh * 2) * 64 + r16];
        v2f b;
        b.x = B0[0];  b.y = B0[64];  c0 = wmma_f32(a, b, c0);
        b.x = B0[16]; b.y = B0[80];  c1 = wmma_f32(a, b, c1);
        b.x = B0[32]; b.y = B0[96];  c2 = wmma_f32(a, b, c2);
        b.x = B0[48]; b.y = B0[112]; c3 = wmma_f32(a, b, c3);
    }
    v8f acc[4] = {c0, c1, c2, c3};
    #pragma unroll
    for (int nt = 0; nt < 4; ++nt) {
        const int col = nt * 16 + r16;
        const float bval = bias[col];
        #pragma unroll
        for (int i = 0; i < 8; ++i) {
            const int row = tile * 16 + kh * 8 + i;
            float val = acc[nt][i] + bval;
            if (MODE == 1) val = gelu_exact(val);
            if (MODE == 3) {
                val += resid[(size_t)row * 64 + col];
                const int b = row >> 12, hw = row & 4095;
                out[((size_t)(b * 64 + col)) * 4096 + hw] = val;   // [B,DIM,H,W]
            } else {
                out[(size_t)row * 64 + col] = val;
            }
        }
    }
}

// ---------------------------------------------------------------------------
// LayerNorm over 64 (+ optional linear sine position embedding).
// One wave per token (lane holds elements d and d+32), 8 tokens per block.
// ---------------------------------------------------------------------------
__device__ __forceinline__ float pos_emb(int d, float cx, float cy) {
    const int grp = d >> 4, idx = d & 15;
    const float f = (float)idx * (3.14f / 200.0f);
    if (grp == 0) return sinf(cx * f);
    if (grp == 1) return cosf(cx * f);
    if (grp == 2) return sinf(cy * f);
    return cosf(cy * f);
}

__global__ void __launch_bounds__(256)
ln_kernel(const float* __restrict__ X, const float* __restrict__ g,
          const float* __restrict__ b, const float* __restrict__ coords1,
          float* __restrict__ out, int withEnc)
{
    const int wave = threadIdx.x >> 5, lane = threadIdx.x & 31;
    const size_t token = (size_t)blockIdx.x * 8 + wave;
    const float* row = X + token * 64;
    const float x0 = row[lane], x1 = row[lane + 32];
    float s = x0 + x1;
    #pragma unroll
    for (int off = 16; off > 0; off >>= 1) s += __shfl_xor(s, off);
    const float mean = s * (1.0f / 64.0f);
    const float d0 = x0 - mean, d1 = x1 - mean;
    float vv = d0 * d0 + d1 * d1;
    #pragma unroll
    for (int off = 16; off > 0; off >>= 1) vv += __shfl_xor(vv, off);
    const float inv = rsqrtf(vv * (1.0f / 64.0f) + 1e-5f);
    float y0 = d0 * inv * g[lane] + b[lane];
    float y1 = d1 * inv * g[lane + 32] + b[lane + 32];
    if (withEnc) {
        const int bs = (int)(token >> 12);
        const int hw = (int)(token & 4095);
        const float cx = coords1[((size_t)bs * 2 + 0) * 4096 + hw];
        const float cy = coords1[((size_t)bs * 2 + 1) * 4096 + hw];
        y0 += pos_emb(lane, cx, cy);
        y1 += pos_emb(lane + 32, cx, cy);
    }
    out[token * 64 + lane] = y0;
    out[token * 64 + lane + 32] = y1;
}

// ---------------------------------------------------------------------------
// Fused attention: k/v = cost_memory @ wk/wv (+bias) via WMMA (waves 0-3: K,
// waves 4-7: V; 2 tokens per 16-row tile), softmax attention (8 keys, 8 heads),
// concat-projection wp + residual. Block = 8 tokens.
// ---------------------------------------------------------------------------
__global__ void __launch_bounds__(256)
attn_kernel(const float* __restrict__ cm,
            const float* __restrict__ wk, const float* __restrict__ bk,
            const float* __restrict__ wv, const float* __restrict__ bv,
            const float* __restrict__ qlin, const float* __restrict__ shortq,
            const float* __restrict__ wp, const float* __restrict__ bp,
            float* __restrict__ xout)
{
    __shared__ float sK[64 * 64];
    __shared__ float sV[64 * 64];
    __shared__ float sOut[8 * 64];
    __shared__ float sShort[8 * 64];
    const int tid = threadIdx.x;
    const int wave = tid >> 5, lane = tid & 31;
    const int r16 = lane & 15, kh = lane >> 4;
    const int wsub = wave & 3;
    const bool isV = wave >= 4;                  // wave-uniform
    const float* W = isV ? wv : wk;
    const float* bias = isV ? bv : bk;
    float* sD = isV ? sV : sK;

    const int localRow = wsub * 16 + r16;        // 0..63 within block's 8 tokens
    const float* Arow = cm + ((size_t)blockIdx.x * 64 + localRow) * 128;
    v8f c0 = {}, c1 = {}, c2 = {}, c3 = {};
    #pragma unroll 2
    for (int kk = 0; kk < 128; kk += 4) {
        v2f a;
        a.x = Arow[kk + kh * 2];
        a.y = Arow[kk + kh * 2 + 1];
        const float* B0 = &W[(kk + kh * 2) * 64 + r16];
        v2f b;
        b.x = B0[0];  b.y = B0[64];  c0 = wmma_f32(a, b, c0);
        b.x = B0[16]; b.y = B0[80];  c1 = wmma_f32(a, b, c1);
        b.x = B0[32]; b.y = B0[96];  c2 = wmma_f32(a, b, c2);
        b.x = B0[48]; b.y = B0[112]; c3 = wmma_f32(a, b, c3);
    }
    v8f acc[4] = {c0, c1, c2, c3};
    #pragma unroll
    for (int nt = 0; nt < 4; ++nt) {
        const int col = nt * 16 + r16;
        const float bval = bias[col];
        #pragma unroll
        for (int i = 0; i < 8; ++i) {
            const int lr = wsub * 16 + kh * 8 + i;
            sD[lr * 64 + col] = acc[nt][i] + bval;
        }
    }
    for (int i = tid; i < 512; i += 256)
        sShort[i] = shortq[(size_t)blockIdx.x * 512 + i];
    __syncthreads();

    if (tid < 64) {                               // one (token,head) per thread
        const int tl = tid >> 3, head = tid & 7;
        const size_t tg = (size_t)blockIdx.x * 8 + tl;
        float q[8];
        #pragma unroll
        for (int d = 0; d < 8; ++d) q[d] = qlin[tg * 64 + head * 8 + d];
        float s[8]; float mx = -1e30f;
        #pragma unroll
        for (int j = 0; j < 8; ++j) {
            const float* kr = &sK[(tl * 8 + j) * 64 + head * 8];
            float dp = 0.0f;
            #pragma unroll
            for (int d = 0; d < 8; ++d) dp += q[d] * kr[d];
            s[j] = dp * 0.35355339059327373f;     // 1/sqrt(8)
            mx = fmaxf(mx, s[j]);
        }
        float den = 0.0f;
        #pragma unroll
        for (int j = 0; j < 8; ++j) { s[j] = expf(s[j] - mx); den += s[j]; }
        const float inv = 1.0f / den;
        #pragma unroll
        for (int d = 0; d < 8; ++d) {
            float o = 0.0f;
            #pragma unroll
            for (int j = 0; j < 8; ++j) o += s[j] * sV[(tl * 8 + j) * 64 + head * 8 + d];
            sOut[tl * 64 + head * 8 + d] = o * inv;
        }
    }
    __syncthreads();

    for (int p = tid; p < 512; p += 256) {        // x = short + [out|short]@wp + bp
        const int tl = p >> 6, col = p & 63;
        float a2 = bp[col];
        const float* o = &sOut[tl * 64];
        const float* sh = &sShort[tl * 64];
        #pragma unroll 8
        for (int k = 0; k < 64; ++k) a2 += o[k] * wp[k * 64 + col];
        #pragma unroll 8
        for (int k = 0; k < 64; ++k) a2 += sh[k] * wp[(64 + k) * 64 + col];
        xout[((size_t)blockIdx.x * 8 + tl) * 64 + col] = sh[col] + a2;
    }
}

// ---------------------------------------------------------------------------
extern "C" void kernel_launch(void* const* d_in, const int* in_sizes, int n_in,
                              void* d_out, int out_size, void* d_ws, size_t ws_size,
                              hipStream_t stream)
{
    const float* cost_maps   = (const float*)d_in[0];
    const float* cost_memory = (const float*)d_in[1];
    const float* coords1     = (const float*)d_in[2];
    const float* up_mask     = (const float*)d_in[3];
    const float* fte_w1 = (const float*)d_in[4];
    const float* fte_b1 = (const float*)d_in[5];
    const float* fte_w2 = (const float*)d_in[6];
    const float* fte_b2 = (const float*)d_in[7];
    const float* ln1_g  = (const float*)d_in[8];
    const float* ln1_b  = (const float*)d_in[9];
    const float* ln2_g  = (const float*)d_in[10];
    const float* ln2_b  = (const float*)d_in[11];
    const float* wq = (const float*)d_in[12];
    const float* bq = (const float*)d_in[13];
    const float* wk = (const float*)d_in[14];
    const float* bk = (const float*)d_in[15];
    const float* wv = (const float*)d_in[16];
    const float* bv = (const float*)d_in[17];
    const float* wp = (const float*)d_in[18];
    const float* bp = (const float*)d_in[19];
    const float* ffn_w1 = (const float*)d_in[20];
    const float* ffn_b1 = (const float*)d_in[21];
    const float* ffn_w2 = (const float*)d_in[22];
    const float* ffn_b2 = (const float*)d_in[23];

    float* out = (float*)d_out;                 // [flow_up: 1048576][cost_global: 524288]
    float* ws  = (float*)d_ws;
    float* corr = ws;                           // 8192*84
    float* q1   = ws + 688128;                  // 8192*64 each below
    float* q2   = ws + 1212416;
    float* qa   = ws + 1736704;
    float* qlin = ws + 2260992;
    float* xbuf = ws + 2785280;
    float* xn   = ws + 3309568;
    float* f1   = ws + 3833856;

    upsample_kernel<<<8192, 64, 0, stream>>>(coords1, up_mask, out);
    corr_kernel<<<(NTOK * 84 + 255) / 256, 256, 0, stream>>>(cost_maps, coords1, corr);
    gemm_n64<84, 81, true, 1><<<64, 256, 0, stream>>>(corr, fte_w1, fte_b1, nullptr, q1);
    gemm_n64<64, 64, true, 0><<<64, 256, 0, stream>>>(q1, fte_w2, fte_b2, nullptr, q2);
    ln_kernel<<<1024, 256, 0, stream>>>(q2, ln1_g, ln1_b, coords1, qa, 1);
    gemm_n64<64, 64, false, 0><<<64, 256, 0, stream>>>(qa, wq, bq, nullptr, qlin);
    attn_kernel<<<1024, 256, 0, stream>>>(cost_memory, wk, bk, wv, bv, qlin, q2, wp, bp, xbuf);
    ln_kernel<<<1024, 256, 0, stream>>>(xbuf, ln2_g, ln2_b, nullptr, xn, 0);
    gemm_n64<64, 64, false, 1><<<64, 256, 0, stream>>>(xn, ffn_w1, ffn_b1, nullptr, f1);
    gemm_n64<64, 64, false, 3><<<64, 256, 0, stream>>>(f1, ffn_w2, ffn_b2, xbuf, out + 1048576);
}